// UNetGAT_20151986553227
// MI455X (gfx1250) — compile-verified
//
#include <hip/hip_runtime.h>
#include <math.h>

#define NN 20000
#define EE 320000
#define EP (EE + NN)
#define NEG_SLOPE 0.2f
#define GEMM_GX 80

typedef __attribute__((ext_vector_type(2))) float v2f;
typedef __attribute__((ext_vector_type(8))) float v8f;

// ---------------------------------------------------------------------------
// float atomic max via integer ordering trick (works for mixed signs)
// ---------------------------------------------------------------------------
__device__ __forceinline__ void atomicMaxF(float* addr, float v) {
  if (v >= 0.0f) atomicMax((int*)addr, __float_as_int(v));
  else           atomicMin((unsigned int*)addr, __float_as_uint(v));
}

// ---------------------------------------------------------------------------
// generic fill
// ---------------------------------------------------------------------------
__global__ void fill_f32(float* __restrict__ p, float v, int n) {
  int i = blockIdx.x * blockDim.x + threadIdx.x;
  int stride = gridDim.x * blockDim.x;
  for (; i < n; i += stride) p[i] = v;
}

// ---------------------------------------------------------------------------
// graph prep: edge list with self loops, degrees, dinv, per-edge norm
// ---------------------------------------------------------------------------
__global__ void build_edges(const int* __restrict__ src, const int* __restrict__ dst,
                            int* __restrict__ es, int* __restrict__ ed) {
  int i = blockIdx.x * blockDim.x + threadIdx.x;
  if (i < EE) { es[i] = src[i]; ed[i] = dst[i]; }
  if (i < NN) { es[EE + i] = i; ed[EE + i] = i; }
}

__global__ void deg_accum(const int* __restrict__ ed, float* __restrict__ deg) {
  int e = blockIdx.x * blockDim.x + threadIdx.x;
  if (e < EP) atomicAdd(&deg[ed[e]], 1.0f);
}

__global__ void make_dinv(float* __restrict__ deg) {
  int i = blockIdx.x * blockDim.x + threadIdx.x;
  if (i < NN) { float d = deg[i]; deg[i] = d > 0.0f ? rsqrtf(d) : 0.0f; }
}

__global__ void make_norm(const int* __restrict__ es, const int* __restrict__ ed,
                          const float* __restrict__ dinv, float* __restrict__ norm) {
  int e = blockIdx.x * blockDim.x + threadIdx.x;
  if (e < EP) norm[e] = dinv[es[e]] * dinv[ed[e]];
}

// ---------------------------------------------------------------------------
// WMMA fp32 GEMM: Out[N x Cout] = A[N x Cin] @ W[Cin x Cout] (+bias)(+relu)
//
// Block = 64 threads (2 waves), grid = (GEMM_GX, ceil(Cout/16)).
// Block owns one 16-column panel of W, staged once into LDS (zero-padded past
// Cout so the k-loop needs NO column guard), pair-interleaved so each lane's
// two K values are one ds_load_b64:
//     Wlds[p*32 + m*2 + j] = W[2p + j][col0 + m]        (j = 0,1)
// Inner loop per k-step of 4: global_load_b64 (A) + ds_load_b64 (B) + WMMA.
// Each wave grid-strides over the 1250 row tiles (20000 = 625 * 32).
// WMMA sits in wave-uniform control flow -> EXEC all ones as required.
// ---------------------------------------------------------------------------
__global__ void __launch_bounds__(64)
gemm_wmma_f32(const float* __restrict__ A, const float* __restrict__ W,
              const float* __restrict__ bias, float* __restrict__ Out,
              int Cin, int Cout, int fuse_relu) {
  __shared__ float Wlds[8192];                 // up to Cin=512 x 16 cols (32 KB)
  const int tid  = threadIdx.x;
  const int col0 = blockIdx.y * 16;

  // ---- stage W panel (pair-interleaved, zero-padded) ----
  const int nwords = Cin * 16;
  for (int idx = tid; idx < nwords; idx += 64) {
    int p  = idx >> 5;          // K pair
    int r  = idx & 31;
    int mm = r >> 1;            // column within panel
    int j  = r & 1;             // K within pair
    int col = col0 + mm;
    Wlds[idx] = (col < Cout) ? W[(size_t)(2 * p + j) * Cout + col] : 0.0f;
  }
  __syncthreads();

  const int lane = tid & 31;
  const int wv   = tid >> 5;
  const int m    = lane & 15;
  const int koff = (lane >> 4) * 2;            // lanes 0-15: K 0,1 ; 16-31: K 2,3
  const int col  = col0 + m;
  const bool colok = (col < Cout);
  const float bv = (bias != nullptr && colok) ? bias[col] : 0.0f;
  const float* bbase = Wlds + (lane >> 4) * 32 + m * 2;

  for (int rt = blockIdx.x * 2 + wv; rt < NN / 16; rt += gridDim.x * 2) {
    const float* ap = A + (size_t)(rt * 16 + m) * Cin + koff;
    const float* bp = bbase;
    v8f acc = {};
    for (int k = 0; k < Cin; k += 4) {
      v2f a = *(const v2f*)ap;                 // A[row][k+koff .. +1]
      v2f b = *(const v2f*)bp;                 // W[k+koff][col], W[k+koff+1][col]
      acc = __builtin_amdgcn_wmma_f32_16x16x4_f32(
          false, a, false, b, (short)0, acc, false, false);
      ap += 4;
      bp += 64;                                // next K pair-of-pairs
    }
    if (colok) {
      const int rbase = rt * 16 + (lane >> 4) * 8;
#pragma unroll
      for (int v = 0; v < 8; ++v) {
        float xv = acc[v] + bv;
        if (fuse_relu) xv = fmaxf(xv, 0.0f);
        Out[(size_t)(rbase + v) * Cout + col] = xv;
      }
    }
  }
}

// ---------------------------------------------------------------------------
// GCN aggregation: out[ed] += h[es] * norm[e]
// thread handles one (edge, 4-channel chunk); C/4 is a power of two.
// ---------------------------------------------------------------------------
__global__ void gcn_scatter4(const float4* __restrict__ h4, const int* __restrict__ es,
                             const int* __restrict__ ed, const float* __restrict__ norm,
                             float* __restrict__ out, int c4shift) {
  int t = blockIdx.x * blockDim.x + threadIdx.x;
  int e = t >> c4shift;
  if (e >= EP) return;
  int q  = t & ((1 << c4shift) - 1);
  int C4 = 1 << c4shift;
  int s = es[e], d = ed[e];
  float w = norm[e];
  float4 v = h4[(size_t)s * C4 + q];
  float* o = out + (((size_t)d * C4 + q) << 2);
  atomicAdd(o + 0, v.x * w);
  atomicAdd(o + 1, v.y * w);
  atomicAdd(o + 2, v.z * w);
  atomicAdd(o + 3, v.w * w);
}

// out = relu(agg + b) (+ skip)
__global__ void layer_finalize(const float* __restrict__ agg, const float* __restrict__ b,
                               const float* __restrict__ skip, float* __restrict__ out, int C) {
  int i = blockIdx.x, c = threadIdx.x;
  size_t idx = (size_t)i * C + c;
  float v = fmaxf(agg[idx] + b[c], 0.0f);
  if (skip) v += skip[idx];
  out[idx] = v;
}

// ---------------------------------------------------------------------------
// GAT pieces
// ---------------------------------------------------------------------------
__global__ void gat_scores(const float* __restrict__ h, const float* __restrict__ att_s,
                           const float* __restrict__ att_d, float* __restrict__ a_s,
                           float* __restrict__ a_d, int H, int C) {
  int t = blockIdx.x * blockDim.x + threadIdx.x;
  if (t >= NN * H) return;
  int n = t / H, hh = t - n * H;
  const float* hp = h + (size_t)n * H * C + (size_t)hh * C;
  const float* ps = att_s + (size_t)hh * C;
  const float* pd = att_d + (size_t)hh * C;
  float ss = 0.0f, sd = 0.0f;
  for (int c = 0; c < C; ++c) { float v = hp[c]; ss += v * ps[c]; sd += v * pd[c]; }
  a_s[t] = ss; a_d[t] = sd;
}

__device__ __forceinline__ float leaky(float x) {
  return x >= 0.0f ? x : x * NEG_SLOPE;
}

__global__ void gat_max(const int* __restrict__ es, const int* __restrict__ ed,
                        const float* __restrict__ a_s, const float* __restrict__ a_d,
                        float* __restrict__ mx, int H) {
  int t = blockIdx.x * blockDim.x + threadIdx.x;
  if (t >= EP * H) return;
  int e = t / H, hh = t - e * H;
  float lg = leaky(a_s[es[e] * H + hh] + a_d[ed[e] * H + hh]);
  atomicMaxF(&mx[ed[e] * H + hh], lg);
}

__global__ void gat_denom(const int* __restrict__ es, const int* __restrict__ ed,
                          const float* __restrict__ a_s, const float* __restrict__ a_d,
                          const float* __restrict__ mx, float* __restrict__ den, int H) {
  int t = blockIdx.x * blockDim.x + threadIdx.x;
  if (t >= EP * H) return;
  int e = t / H, hh = t - e * H;
  int d = ed[e] * H + hh;
  float lg = leaky(a_s[es[e] * H + hh] + a_d[d]);
  atomicAdd(&den[d], expf(lg - mx[d]));
}

// thread handles one (edge, 4-channel chunk) of the H*C feature vector;
// hc4shift = log2(H*C/4), cshift = log2(C); alpha recomputed per thread.
__global__ void gat_scatter4(const float4* __restrict__ h4, const int* __restrict__ es,
                             const int* __restrict__ ed, const float* __restrict__ a_s,
                             const float* __restrict__ a_d, const float* __restrict__ mx,
                             const float* __restrict__ den, float* __restrict__ out,
                             int H, int hc4shift, int cshift) {
  int t = blockIdx.x * blockDim.x + threadIdx.x;
  int e = t >> hc4shift;
  if (e >= EP) return;
  int q   = t & ((1 << hc4shift) - 1);       // 4-chunk index within H*C
  int HC4 = 1 << hc4shift;
  int hh  = (q << 2) >> cshift;              // head index
  int s = es[e], d = ed[e];
  int sd = d * H + hh;
  float lg = leaky(a_s[s * H + hh] + a_d[sd]);
  float alpha = expf(lg - mx[sd]) / den[sd];
  float4 v = h4[(size_t)s * HC4 + q];
  float* o = out + (((size_t)d * HC4 + q) << 2);
  atomicAdd(o + 0, v.x * alpha);
  atomicAdd(o + 1, v.y * alpha);
  atomicAdd(o + 2, v.z * alpha);
  atomicAdd(o + 3, v.w * alpha);
}

// ---------------------------------------------------------------------------
// host-side layer helpers (launches only; all on `stream`)
// ---------------------------------------------------------------------------
static void run_gemm(const float* A, const float* W, const float* bias, float* Out,
                     int cin, int cout, int relu, hipStream_t s) {
  dim3 grid(GEMM_GX, (cout + 15) / 16);
  gemm_wmma_f32<<<grid, dim3(64), 0, s>>>(A, W, bias, Out, cin, cout, relu);
}

static void gcn_layer(const float* X, const float* W, const float* b, int cin, int cout,
                      const int* es, const int* ed, const float* norm,
                      float* hbuf, float* abuf, const float* skip, float* out,
                      hipStream_t s) {
  run_gemm(X, W, nullptr, hbuf, cin, cout, 0, s);
  int total = NN * cout;
  fill_f32<<<(total + 255) / 256, 256, 0, s>>>(abuf, 0.0f, total);
  int c4shift = __builtin_ctz(cout >> 2);
  long long work = (long long)EP << c4shift;
  gcn_scatter4<<<(unsigned)((work + 255) / 256), 256, 0, s>>>(
      (const float4*)hbuf, es, ed, norm, abuf, c4shift);
  layer_finalize<<<NN, cout, 0, s>>>(abuf, b, skip, out, cout);
}

static void gat_layer(const float* X, const float* W, const float* as, const float* ad,
                      const float* b, int cin, int H, int C,
                      const int* es, const int* ed,
                      float* hbuf, float* abuf, float* a_s, float* a_d, float* mx,
                      float* den, float* out, hipStream_t s) {
  int HC = H * C;
  run_gemm(X, W, nullptr, hbuf, cin, HC, 0, s);
  gat_scores<<<(NN * H + 255) / 256, 256, 0, s>>>(hbuf, as, ad, a_s, a_d, H, C);
  fill_f32<<<(NN * H + 255) / 256, 256, 0, s>>>(mx, -3.0e38f, NN * H);
  gat_max<<<(EP * H + 255) / 256, 256, 0, s>>>(es, ed, a_s, a_d, mx, H);
  fill_f32<<<(NN * H + 255) / 256, 256, 0, s>>>(den, 0.0f, NN * H);
  gat_denom<<<(EP * H + 255) / 256, 256, 0, s>>>(es, ed, a_s, a_d, mx, den, H);
  fill_f32<<<(NN * HC + 255) / 256, 256, 0, s>>>(abuf, 0.0f, NN * HC);
  int hc4shift = __builtin_ctz(HC >> 2);
  int cshift   = __builtin_ctz(C);
  long long work = (long long)EP << hc4shift;
  gat_scatter4<<<(unsigned)((work + 255) / 256), 256, 0, s>>>(
      (const float4*)hbuf, es, ed, a_s, a_d, mx, den, abuf, H, hc4shift, cshift);
  layer_finalize<<<NN, HC, 0, s>>>(abuf, b, nullptr, out, HC);
}

// ---------------------------------------------------------------------------
// entry point
// ---------------------------------------------------------------------------
extern "C" void kernel_launch(void* const* d_in, const int* in_sizes, int n_in,
                              void* d_out, int out_size, void* d_ws, size_t ws_size,
                              hipStream_t stream) {
  (void)in_sizes; (void)n_in; (void)out_size; (void)ws_size;

  const float* x     = (const float*)d_in[0];
  const int*   ei    = (const int*)  d_in[1];     // [2, E]: src then dst
  const float* w_e1  = (const float*)d_in[2];  const float* b_e1 = (const float*)d_in[3];
  const float* w_e2  = (const float*)d_in[4];  const float* b_e2 = (const float*)d_in[5];
  const float* w_e3  = (const float*)d_in[6];
  const float* as_e3 = (const float*)d_in[7];  const float* ad_e3 = (const float*)d_in[8];
  const float* b_e3  = (const float*)d_in[9];
  const float* w_d1  = (const float*)d_in[10]; const float* b_d1 = (const float*)d_in[11];
  const float* w_d2  = (const float*)d_in[12]; const float* b_d2 = (const float*)d_in[13];
  const float* w_d3  = (const float*)d_in[14];
  const float* as_d3 = (const float*)d_in[15]; const float* ad_d3 = (const float*)d_in[16];
  const float* b_d3  = (const float*)d_in[17];
  const float* w_fc1 = (const float*)d_in[18]; const float* b_fc1 = (const float*)d_in[19];
  const float* w_fc2 = (const float*)d_in[20]; const float* b_fc2 = (const float*)d_in[21];
  const float* w_fin = (const float*)d_in[22]; const float* b_fin = (const float*)d_in[23];

  float* ws = (float*)d_ws;
  int*   es   = (int*)(ws + 0);            // 340000 ints
  int*   ed   = (int*)(ws + 340000);       // 340000 ints
  float* dinv = ws + 680000;               // 20000
  float* norm = ws + 700000;               // 340000
  float* a_s  = ws + 1040000;              // 80000
  float* a_d  = ws + 1120000;              // 80000
  float* mx   = ws + 1200000;              // 80000
  float* den  = ws + 1280000;              // 80000
  float* x1   = ws + 1360000;              // 20000*64
  float* x2   = ws + 2640000;              // 20000*128
  float* x3   = ws + 5200000;              // 20000*512
  float* hbuf = ws + 15440000;             // 20000*512 (gemm scratch)
  float* abuf = ws + 25680000;             // 20000*512 (agg scratch)
  float* t1   = ws + 35920000;             // 20000*128
  float* t2   = ws + 38480000;             // 20000*64
  float* t3   = ws + 39760000;             // 20000*256

  // ---- graph prep -------------------------------------------------------
  build_edges<<<(EE + 255) / 256, 256, 0, stream>>>(ei, ei + EE, es, ed);
  fill_f32<<<(NN + 255) / 256, 256, 0, stream>>>(dinv, 0.0f, NN);
  deg_accum<<<(EP + 255) / 256, 256, 0, stream>>>(ed, dinv);
  make_dinv<<<(NN + 255) / 256, 256, 0, stream>>>(dinv);
  make_norm<<<(EP + 255) / 256, 256, 0, stream>>>(es, ed, dinv, norm);

  // ---- encoder ----------------------------------------------------------
  gcn_layer(x,  w_e1, b_e1, 32,  64,  es, ed, norm, hbuf, abuf, nullptr, x1, stream);
  gcn_layer(x1, w_e2, b_e2, 64,  128, es, ed, norm, hbuf, abuf, nullptr, x2, stream);
  gat_layer(x2, w_e3, as_e3, ad_e3, b_e3, 128, 4, 128,
            es, ed, hbuf, abuf, a_s, a_d, mx, den, x3, stream);

  // ---- decoder ----------------------------------------------------------
  gcn_layer(x3, w_d1, b_d1, 512, 128, es, ed, norm, hbuf, abuf, x2, t1, stream);
  gcn_layer(t1, w_d2, b_d2, 128, 64,  es, ed, norm, hbuf, abuf, x1, t2, stream);
  gat_layer(t2, w_d3, as_d3, ad_d3, b_d3, 64, 4, 64,
            es, ed, hbuf, abuf, a_s, a_d, mx, den, t3, stream);

  // ---- MLP head (WMMA GEMMs, fused bias+relu) ---------------------------
  run_gemm(t3,   w_fc1, b_fc1, hbuf, 256, 256, 1, stream);   // 256 -> 256, relu
  run_gemm(hbuf, w_fc2, b_fc2, abuf, 256, 128, 1, stream);   // 256 -> 128, relu
  run_gemm(abuf, w_fin, b_fin, (float*)d_out, 128, 4, 0, stream); // 128 -> 4
}